// meta_predictor_end2end_1090921693494
// MI455X (gfx1250) — compile-verified
//
#include <hip/hip_runtime.h>
#include <math.h>

typedef float v2f __attribute__((ext_vector_type(2)));
typedef float v8f __attribute__((ext_vector_type(8)));

#define B_    32
#define N_    1024
#define D_    32
#define M_    32
#define NCOL  10
#define VOCAB 8
#define EDIM  3
#define NCHUNK 16                 // N-chunks per (b, dtile)
#define CHUNK_N (N_ / NCHUNK)     // 64 n per wave

// One-instruction relu. fmaxf / fmed3 both get lowered by clang to a 2-op
// canonicalize+max sequence; pin the single v_max_num_f32 with inline asm.
__device__ __forceinline__ float relu1(float x) {
    float r;
    asm("v_max_num_f32 %0, 0, %1" : "=v"(r) : "v"(x));
    return r;
}

// ---------------------------------------------------------------------------
// Kernel 1: f_feat partial sums.
//   f[b,n,d,m] = relu(X[b,n,d]*Wf[0,m] + y[b,n]*Wf[1,m] + bf[m])
//   partial[b][dtile][chunk][d_local][m] = sum over this wave's 64 n of f
// WMMA tile: rows = 16 d-values (fixed b,n), K = [x, y, 0, 0], cols = 16 m's.
// A layout (32-bit 16x4): lanes 0-15 hold K=0 (VGPR0) / K=1 (VGPR1); lanes
// 16-31 hold the K=2/3 padding, forced to zero so padding contributes nothing.
// Bias is carried in C; relu + accumulate must happen per n (mean of relu).
// ---------------------------------------------------------------------------
__global__ __launch_bounds__(128)
void k_ffeat(const float* __restrict__ X, const float* __restrict__ Y,
             const float* __restrict__ Wf, const float* __restrict__ bf,
             float* __restrict__ partial)
{
    const int lane  = threadIdx.x & 31;
    const int wave  = threadIdx.x >> 5;
    const int b     = blockIdx.x >> 3;          // 32 b
    const int rem   = blockIdx.x & 7;
    const int dtile = rem >> 2;                 // 2 d-tiles
    const int chunk = (rem & 3) * 4 + wave;     // 16 chunks
    const int nbase = chunk * CHUNK_N;
    const int col   = lane & 15;
    const bool lo   = lane < 16;

    // B matrices (4x16): K rows 0/1 = Wf rows; K rows 2/3 = zero padding.
    v2f wb0, wb1;
    wb0.x = lo ? Wf[0 * M_ + 0 * 16 + col] : 0.f;   // K=0 (x weight), m-tile 0
    wb0.y = lo ? Wf[1 * M_ + 0 * 16 + col] : 0.f;   // K=1 (y weight)
    wb1.x = lo ? Wf[0 * M_ + 1 * 16 + col] : 0.f;   // m-tile 1
    wb1.y = lo ? Wf[1 * M_ + 1 * 16 + col] : 0.f;

    // Bias in C (broadcast down each column).
    const float bv0 = bf[0 * 16 + col];
    const float bv1 = bf[1 * 16 + col];
    v8f c0 = {bv0, bv0, bv0, bv0, bv0, bv0, bv0, bv0};
    v8f c1 = {bv1, bv1, bv1, bv1, bv1, bv1, bv1, bv1};

    v8f acc0 = {0.f, 0.f, 0.f, 0.f, 0.f, 0.f, 0.f, 0.f};
    v8f acc1 = {0.f, 0.f, 0.f, 0.f, 0.f, 0.f, 0.f, 0.f};

    const float* Xb = X + (size_t)b * N_ * D_ + dtile * 16;
    const float* Yb = Y + (size_t)b * N_;

    for (int i = 0; i < CHUNK_N; ++i) {
        const int n = nbase + i;
        const float xv = Xb[(size_t)n * D_ + col];   // coalesced; upper half dup
        const float yv = Yb[n];                      // wave-uniform
        v2f a;
        a.x = lo ? xv : 0.f;                         // K=0 | K=2 padding
        a.y = lo ? yv : 0.f;                         // K=1 | K=3 padding
        v8f d0 = __builtin_amdgcn_wmma_f32_16x16x4_f32(
            false, a, false, wb0, (short)0, c0, false, false);
        v8f d1 = __builtin_amdgcn_wmma_f32_16x16x4_f32(
            false, a, false, wb1, (short)0, c1, false, false);
#pragma unroll
        for (int j = 0; j < 8; ++j) {
            acc0[j] += relu1(d0[j]);
            acc1[j] += relu1(d1[j]);
        }
    }

    // Disjoint partial slot per wave -> deterministic (no float atomics).
    float* p = partial + ((((size_t)b * 2 + dtile) * NCHUNK + chunk) * 16) * 32;
    const int rhalf = (lane >> 4) * 8;   // C/D layout: lanes 16-31 hold M=8..15
#pragma unroll
    for (int j = 0; j < 8; ++j) {
        const int dl = j + rhalf;
        p[dl * 32 + 0 * 16 + col] = acc0[j];
        p[dl * 32 + 1 * 16 + col] = acc1[j];
    }
}

// ---------------------------------------------------------------------------
// Kernel 2: reduce partials (mean over N), then g -> mean over D -> h ->
// embedding concat -> classifier -> sigmoid. One block; latency-bound tail.
// ---------------------------------------------------------------------------
__global__ __launch_bounds__(1024)
void k_tail(const float* __restrict__ partial,
            const float* __restrict__ nla, const int* __restrict__ comp,
            const float* __restrict__ Wg1, const float* __restrict__ bg1,
            const float* __restrict__ Wg2, const float* __restrict__ bg2,
            const float* __restrict__ Wh1, const float* __restrict__ bh1,
            const float* __restrict__ Wh2, const float* __restrict__ bh2,
            const float* __restrict__ emb_tables,
            const float* __restrict__ Wc1, const float* __restrict__ bc1,
            const float* __restrict__ Wc2, const float* __restrict__ bc2,
            float* __restrict__ out)
{
    __shared__ float s_g2[B_ * D_ * 8];   // 32 KB
    __shared__ float s_gf[B_ * 8];

    const int t = threadIdx.x;

    // Stage A: per (b,d) row -> f_feat -> g1 -> g2
    {
        const int b = t >> 5, d = t & 31;
        const int dtile = d >> 4, dl = d & 15;
        float ff[32];
#pragma unroll
        for (int k = 0; k < 32; ++k) ff[k] = 0.f;
        const float* p = partial +
            (((size_t)b * 2 + dtile) * NCHUNK * 16 + dl) * 32;
        for (int c = 0; c < NCHUNK; ++c) {
            const float* pc = p + (size_t)c * 16 * 32;
#pragma unroll
            for (int k = 0; k < 32; ++k) ff[k] += pc[k];
        }
        const float invN = 1.f / (float)N_;
#pragma unroll
        for (int k = 0; k < 32; ++k) ff[k] *= invN;

        float g1[16];
        for (int j = 0; j < 16; ++j) {
            float s = bg1[j];
            for (int k = 0; k < 32; ++k) s += ff[k] * Wg1[k * 16 + j];
            g1[j] = relu1(s);
        }
        for (int j = 0; j < 8; ++j) {
            float s = bg2[j];
            for (int k = 0; k < 16; ++k) s += g1[k] * Wg2[k * 8 + j];
            s_g2[t * 8 + j] = relu1(s);
        }
    }
    __syncthreads();

    // Stage B: mean over D
    if (t < B_ * 8) {
        const int b = t >> 3, j = t & 7;
        float s = 0.f;
        for (int d = 0; d < D_; ++d) s += s_g2[(b * D_ + d) * 8 + j];
        s_gf[b * 8 + j] = s * (1.f / (float)D_);
    }
    __syncthreads();

    // Stage C: h, embedding concat, classifier, sigmoid (one thread per b)
    if (t < B_) {
        const int b = t;
        float h1[16];
        for (int j = 0; j < 16; ++j) {
            float s = bh1[j];
            for (int k = 0; k < 8; ++k) s += s_gf[b * 8 + k] * Wh1[k * 16 + j];
            h1[j] = relu1(s);
        }
        float eb[63];
        for (int j = 0; j < 32; ++j) {
            float s = bh2[j];
            for (int k = 0; k < 16; ++k) s += h1[k] * Wh2[k * 32 + j];
            s = relu1(s);
            out[b * 32 + j] = s;          // output 0: meta_features [32,32]
            eb[j] = s;
        }
        eb[32] = nla[b];
        for (int c = 0; c < NCOL; ++c) {
            const int idx = comp[b * NCOL + c];
            const float* e = emb_tables + ((size_t)c * VOCAB + idx) * EDIM;
            eb[33 + c * 3 + 0] = e[0];
            eb[33 + c * 3 + 1] = e[1];
            eb[33 + c * 3 + 2] = e[2];
        }
        for (int j = 0; j < 63; ++j)      // output 1: embedding [32,63]
            out[B_ * M_ + b * 63 + j] = eb[j];

        float c1r[20];
        for (int j = 0; j < 20; ++j) {
            float s = bc1[j];
            for (int k = 0; k < 63; ++k) s += eb[k] * Wc1[k * 20 + j];
            c1r[j] = relu1(s);
        }
        float s = bc2[0];
        for (int j = 0; j < 20; ++j) s += c1r[j] * Wc2[j];
        out[B_ * M_ + B_ * 63 + b] = 1.f / (1.f + expf(-s));  // output 2: pred
    }
}

extern "C" void kernel_launch(void* const* d_in, const int* in_sizes, int n_in,
                              void* d_out, int out_size, void* d_ws, size_t ws_size,
                              hipStream_t stream) {
    (void)in_sizes; (void)n_in; (void)out_size; (void)ws_size;
    const float* X          = (const float*)d_in[0];
    const float* y          = (const float*)d_in[1];
    const float* nla        = (const float*)d_in[2];
    const int*   components = (const int*)  d_in[3];
    const float* Wf  = (const float*)d_in[4];
    const float* bf  = (const float*)d_in[5];
    const float* Wg1 = (const float*)d_in[6];
    const float* bg1 = (const float*)d_in[7];
    const float* Wg2 = (const float*)d_in[8];
    const float* bg2 = (const float*)d_in[9];
    const float* Wh1 = (const float*)d_in[10];
    const float* bh1 = (const float*)d_in[11];
    const float* Wh2 = (const float*)d_in[12];
    const float* bh2 = (const float*)d_in[13];
    const float* emb = (const float*)d_in[14];
    const float* Wc1 = (const float*)d_in[15];
    const float* bc1 = (const float*)d_in[16];
    const float* Wc2 = (const float*)d_in[17];
    const float* bc2 = (const float*)d_in[18];

    float* partial = (float*)d_ws;   // B*2*NCHUNK*16*32 = 524288 floats = 2 MB
    float* out     = (float*)d_out;  // 1024 + 2016 + 32 = 3072 floats

    k_ffeat<<<dim3(B_ * 2 * (NCHUNK / 4)), dim3(128), 0, stream>>>(
        X, y, Wf, bf, partial);
    k_tail<<<dim3(1), dim3(1024), 0, stream>>>(
        partial, nla, components, Wg1, bg1, Wg2, bg2,
        Wh1, bh1, Wh2, bh2, emb, Wc1, bc1, Wc2, bc2, out);
}